// CenterWeightedCIoULoss_89850715832989
// MI455X (gfx1250) — compile-verified
//
#include <hip/hip_runtime.h>
#include <hip/hip_bf16.h>

// CenterWeightedCIoULoss for MI455X (gfx1250, wave32).
//
// Roofline: 2 * 4,194,304 * 16 B = 134 MB streamed once, scalar out.
// At 23.3 TB/s this is ~5.8 us -> pure bandwidth problem. Strategy:
//   pass 1: 2048 blocks x 256 threads, each thread does 8 iterations of
//           two GLOBAL_LOAD_B128 (one float4 per box), fp32 math, register
//           accumulation. Nontemporal loads (stream-once data).
//   wave reduce: pack each lane's partial into B of V_WMMA_F32_16X16X4_F32
//           with A = ones(16x4). D[i][j] = sum_k B[k][j], so summing a
//           16-lane half of row 0 yields the exact 32-lane sum. 1 WMMA +
//           4 ds_swizzle xor-swaps (single-instruction lane exchange, no
//           bpermute address math) on the matrix/LDS pipes.
//   pass 2: one block folds the 2048 per-block partials the same way and
//           writes mean = sum / N. Fixed reduction tree -> deterministic.

typedef __attribute__((ext_vector_type(2))) float v2f;
typedef __attribute__((ext_vector_type(4))) float v4f;   // native vec: OK for nontemporal builtin
typedef __attribute__((ext_vector_type(8))) float v8f;

#define EPS 1e-7f
#define CENTER_WEIGHT 2.0f

// ds_swizzle_b32 group-of-32 XOR pattern: offset[15]=0, xor_mask=[14:10],
// or_mask=[9:5]=0, and_mask=[4:0]=0x1f.
#define SWIZZLE_XOR(m) (((m) << 10) | 0x1f)

// Exact 32-lane sum via the matrix pipe.
// A = ones(16x4), B holds {partial, 0} per lane (4x16 = 64 slots, each
// (lane, vgpr) slot covered exactly once). D[i][j] = sum_k B[k][j], i.e.
// every row of D is the vector of column sums; lanes 0..15 of d[0] hold
// colsum[0..15] (lanes 16..31 hold the identical M=8 row). Summing a
// 16-lane half with 4 xor-swizzles gives the total in every lane.
// Requires EXEC == all 1s (call only from fully-active full waves).
__device__ __forceinline__ float wave_sum_wmma(float partial) {
    v2f a; a[0] = 1.0f; a[1] = 1.0f;          // ones A-matrix
    v2f b; b[0] = partial; b[1] = 0.0f;       // lane partial into B
    v8f c = {};
    c = __builtin_amdgcn_wmma_f32_16x16x4_f32(
        /*neg_a=*/false, a, /*neg_b=*/false, b,
        /*c_mod=*/(short)0, c, /*reuse_a=*/false, /*reuse_b=*/false);
    float s = c[0];
    s += __int_as_float(__builtin_amdgcn_ds_swizzle(__float_as_int(s), SWIZZLE_XOR(1)));
    s += __int_as_float(__builtin_amdgcn_ds_swizzle(__float_as_int(s), SWIZZLE_XOR(2)));
    s += __int_as_float(__builtin_amdgcn_ds_swizzle(__float_as_int(s), SWIZZLE_XOR(4)));
    s += __int_as_float(__builtin_amdgcn_ds_swizzle(__float_as_int(s), SWIZZLE_XOR(8)));
    return s;   // total 32-lane sum (each 16-lane half independently holds it)
}

__device__ __forceinline__ float ciou_loss(v4f p, v4f t) {
    const float px1 = p[0], py1 = p[1], px2 = p[2], py2 = p[3];
    const float tx1 = t[0], ty1 = t[1], tx2 = t[2], ty2 = t[3];

    float iw = fmaxf(fminf(px2, tx2) - fmaxf(px1, tx1), 0.0f);
    float ih = fmaxf(fminf(py2, ty2) - fmaxf(py1, ty1), 0.0f);
    float inter = iw * ih;
    float pa = (px2 - px1) * (py2 - py1);
    float ta = (tx2 - tx1) * (ty2 - ty1);
    float iou = inter / (pa + ta - inter + EPS);

    float dcx = (px1 + px2) * 0.5f - (tx1 + tx2) * 0.5f;
    float dcy = (py1 + py2) * 0.5f - (ty1 + ty2) * 0.5f;
    float center_dist_sq = dcx * dcx + dcy * dcy;

    float cw = fmaxf(px2, tx2) - fminf(px1, tx1);
    float ch = fmaxf(py2, ty2) - fminf(py1, ty1);
    float c_diag_sq = cw * cw + ch * ch + EPS;          // bbox_iou eps
    float center_term = center_dist_sq / (c_diag_sq + EPS);  // forward() eps again

    float pw = px2 - px1, ph = py2 - py1;
    float tw = tx2 - tx1, th = ty2 - ty1;
    float rw = (pw - tw) / (tw + EPS);
    float rh = (ph - th) / (th + EPS);
    float size_term = rw * rw + rh * rh;

    return (1.0f - iou) + CENTER_WEIGHT * center_term + size_term;
}

__global__ __launch_bounds__(256) void ciou_partial_kernel(
    const v4f* __restrict__ pred, const v4f* __restrict__ tgt,
    float* __restrict__ block_sums, int n) {
    const int tid = blockIdx.x * blockDim.x + threadIdx.x;
    const int stride = gridDim.x * blockDim.x;

    float acc = 0.0f;
    for (int i = tid; i < n; i += stride) {
        // stream-once data: nontemporal (TH=NT) B128 loads
        v4f p = __builtin_nontemporal_load(&pred[i]);
        v4f t = __builtin_nontemporal_load(&tgt[i]);
        acc += ciou_loss(p, t);
    }

    float wsum = wave_sum_wmma(acc);        // matrix-pipe wave reduction

    __shared__ float lds[8];                // 256 threads = 8 wave32
    const int lane = threadIdx.x & 31;
    const int wave = threadIdx.x >> 5;
    if (lane == 0) lds[wave] = wsum;
    __syncthreads();
    if (threadIdx.x == 0) {
        float s = 0.0f;
        const int nwaves = blockDim.x >> 5;
        #pragma unroll
        for (int w = 0; w < 8; ++w)
            if (w < nwaves) s += lds[w];
        block_sums[blockIdx.x] = s;
    }
}

__global__ __launch_bounds__(256) void ciou_final_kernel(
    const float* __restrict__ block_sums, int nblocks,
    float* __restrict__ out, float inv_n) {
    float acc = 0.0f;
    for (int i = threadIdx.x; i < nblocks; i += blockDim.x)
        acc += block_sums[i];

    float wsum = wave_sum_wmma(acc);

    __shared__ float lds[8];
    const int lane = threadIdx.x & 31;
    const int wave = threadIdx.x >> 5;
    if (lane == 0) lds[wave] = wsum;
    __syncthreads();
    if (threadIdx.x == 0) {
        float s = 0.0f;
        const int nwaves = blockDim.x >> 5;
        #pragma unroll
        for (int w = 0; w < 8; ++w)
            if (w < nwaves) s += lds[w];
        out[0] = s * inv_n;
    }
}

extern "C" void kernel_launch(void* const* d_in, const int* in_sizes, int n_in,
                              void* d_out, int out_size, void* d_ws, size_t ws_size,
                              hipStream_t stream) {
    (void)n_in; (void)out_size;
    const v4f* pred = (const v4f*)d_in[0];
    const v4f* tgt  = (const v4f*)d_in[1];
    float* out = (float*)d_out;
    float* block_sums = (float*)d_ws;

    const int n = in_sizes[0] / 4;          // 4 floats per box

    // 2048 blocks x 256 threads (8 wave32/block): 524,288 threads, 8 iters
    // each at N = 4M -> enough in-flight B128 loads to saturate 23.3 TB/s.
    int nblocks = 2048;
    const size_t ws_floats = ws_size / sizeof(float);
    if ((size_t)nblocks > ws_floats) nblocks = (int)ws_floats;
    if (nblocks < 1) nblocks = 1;

    ciou_partial_kernel<<<nblocks, 256, 0, stream>>>(pred, tgt, block_sums, n);
    ciou_final_kernel<<<1, 256, 0, stream>>>(block_sums, nblocks, out,
                                             1.0f / (float)n);
}